// InputTreeAttention_74217034875209
// MI455X (gfx1250) — compile-verified
//
#include <hip/hip_runtime.h>
#include <hip/hip_bf16.h>

typedef __attribute__((ext_vector_type(2))) float v2f;
typedef __attribute__((ext_vector_type(8))) float v8f;

#define HD 1024           // hidden dim
#define NCHUNK 64         // row chunks for weighted sum

// ---------------------------------------------------------------------------
// 0. tiny init: reset the softmax-max accumulator (uint-bit max of positives)
// ---------------------------------------------------------------------------
__global__ void kInit(unsigned* mbits) { *mbits = 0u; }

// ---------------------------------------------------------------------------
// 1. y[0:1024] = W @ x + b   via V_WMMA_F32_16X16X4_F32 (full fp32)
//    W row-major [1024][1024]; one wave per 16-row tile; K-loop in steps of 4.
//    A lane layout: lane L -> row M=L%16, VGPR v -> K = v + 2*(L>=16)
//    B broadcast across N: every lane loads x[k0 + 2*hi .. +1]
// ---------------------------------------------------------------------------
__global__ __launch_bounds__(32) void kMatvecWmma(const float* __restrict__ W,
                                                  const float* __restrict__ x,
                                                  const float* __restrict__ b,
                                                  float* __restrict__ y) {
  const int tile = blockIdx.x;           // 64 tiles of 16 rows
  const int lane = threadIdx.x;          // 0..31
  const int m    = lane & 15;
  const int hi   = lane >> 4;            // 0 or 1
  const int row  = tile * 16 + m;

  const float2* wp = (const float2*)(W + (size_t)row * HD + 2 * hi);
  const float2* xp = (const float2*)(x + 2 * hi);

  v8f acc = {};
  #pragma unroll 8
  for (int kk = 0; kk < HD / 2; kk += 2) {   // 4 K-values per WMMA
    float2 a2 = wp[kk];
    float2 b2 = xp[kk];
    v2f a; a.x = a2.x; a.y = a2.y;
    v2f bb; bb.x = b2.x; bb.y = b2.y;
    acc = __builtin_amdgcn_wmma_f32_16x16x4_f32(false, a, false, bb,
                                                (short)0, acc, false, false);
  }
  // All N columns identical: lanes 0 / 16 hold rows tile*16 + v + 8*hi in acc[v]
  if (m == 0) {
    float*       yo = y + tile * 16 + hi * 8;
    const float* bo = b + tile * 16 + hi * 8;
    #pragma unroll
    for (int v = 0; v < 8; ++v) yo[v] = acc[v] + bo[v];
  }
}

// ---------------------------------------------------------------------------
// 2. s[i] = exp(H[i,:] . q); track global max via uint atomicMax (s > 0).
//    8 waves/block, one row per wave, float4 coalesced loads (512B/wave/iter).
// ---------------------------------------------------------------------------
__global__ __launch_bounds__(256) void kScores(const float* __restrict__ H,
                                               const float* __restrict__ q,
                                               float* __restrict__ s,
                                               unsigned* __restrict__ mbits,
                                               int nrows) {
  const int wave = threadIdx.x >> 5;
  const int lane = threadIdx.x & 31;
  const int row  = blockIdx.x * 8 + wave;
  if (row >= nrows) return;

  const float4* hr = (const float4*)(H + (size_t)row * HD);
  const float4* qv = (const float4*)q;
  float acc = 0.f;
  #pragma unroll
  for (int it = 0; it < HD / (32 * 4); ++it) {
    const int idx = it * 32 + lane;
    float4 h4 = hr[idx];
    float4 q4 = qv[idx];
    acc += h4.x * q4.x + h4.y * q4.y + h4.z * q4.z + h4.w * q4.w;
  }
  #pragma unroll
  for (int off = 16; off > 0; off >>= 1) acc += __shfl_xor(acc, off, 32);
  if (lane == 0) {
    const float sv = expf(acc);
    s[row] = sv;
    atomicMax(mbits, __float_as_uint(sv));   // positive floats: bit order == numeric order
  }
}

// ---------------------------------------------------------------------------
// 3. single block: w[i] = exp(s[i] - m); denom = sum(w)  (fixed-order reduce)
// ---------------------------------------------------------------------------
__global__ __launch_bounds__(1024) void kDenom(const float* __restrict__ s,
                                               float* __restrict__ w,
                                               const unsigned* __restrict__ mbits,
                                               float* __restrict__ denom, int n) {
  __shared__ float red[1024];
  const float m = __uint_as_float(*mbits);
  float acc = 0.f;
  for (int i = threadIdx.x; i < n; i += 1024) {
    const float wi = expf(s[i] - m);
    w[i] = wi;
    acc += wi;
  }
  red[threadIdx.x] = acc;
  __syncthreads();
  for (int st = 512; st > 0; st >>= 1) {
    if (threadIdx.x < st) red[threadIdx.x] += red[threadIdx.x + st];
    __syncthreads();
  }
  if (threadIdx.x == 0) *denom = red[0];
}

// ---------------------------------------------------------------------------
// 4. es_part[chunk][col] = sum over chunk rows of w[i]*H[i][col]
//    grid = NCHUNK row-chunks x 4 col-tiles; disjoint outputs, no atomics.
// ---------------------------------------------------------------------------
__global__ __launch_bounds__(256) void kWeightedSum(const float* __restrict__ H,
                                                    const float* __restrict__ w,
                                                    float* __restrict__ es_part,
                                                    int n, int rowsPerChunk) {
  const int tile  = blockIdx.x & 3;
  const int chunk = blockIdx.x >> 2;
  const int col   = tile * 256 + threadIdx.x;
  const int r0    = chunk * rowsPerChunk;
  const int r1    = min(r0 + rowsPerChunk, n);
  float acc = 0.f;
  for (int i = r0; i < r1; ++i)
    acc = fmaf(w[i], H[(size_t)i * HD + col], acc);   // w[i] uniform -> scalar load
  es_part[chunk * HD + col] = acc;
}

// ---------------------------------------------------------------------------
// 5. es[col] = (sum_c es_part[c][col]) / denom   (fixed order -> deterministic)
// ---------------------------------------------------------------------------
__global__ __launch_bounds__(256) void kReduceEs(const float* __restrict__ es_part,
                                                 const float* __restrict__ denom,
                                                 float* __restrict__ es) {
  const int col = blockIdx.x * 256 + threadIdx.x;
  float acc = 0.f;
  #pragma unroll 8
  for (int c = 0; c < NCHUNK; ++c) acc += es_part[c * HD + col];
  es[col] = acc / (*denom);
}

// ---------------------------------------------------------------------------
// 6. out = tanh(W1 @ es + b1 + W2 @ h_t + b2)  — fused double WMMA matvec
// ---------------------------------------------------------------------------
__global__ __launch_bounds__(32) void kFinalWmma(const float* __restrict__ W1,
                                                 const float* __restrict__ b1,
                                                 const float* __restrict__ W2,
                                                 const float* __restrict__ b2,
                                                 const float* __restrict__ es,
                                                 const float* __restrict__ ht,
                                                 float* __restrict__ out) {
  const int tile = blockIdx.x;
  const int lane = threadIdx.x;
  const int m    = lane & 15;
  const int hi   = lane >> 4;
  const int row  = tile * 16 + m;

  const float2* w1p = (const float2*)(W1 + (size_t)row * HD + 2 * hi);
  const float2* w2p = (const float2*)(W2 + (size_t)row * HD + 2 * hi);
  const float2* xp  = (const float2*)(es + 2 * hi);
  const float2* hp  = (const float2*)(ht + 2 * hi);

  v8f acc = {};
  #pragma unroll 4
  for (int kk = 0; kk < HD / 2; kk += 2) {
    float2 a2 = w1p[kk], b2v = xp[kk];
    v2f a; a.x = a2.x; a.y = a2.y;
    v2f bb; bb.x = b2v.x; bb.y = b2v.y;
    acc = __builtin_amdgcn_wmma_f32_16x16x4_f32(false, a, false, bb,
                                                (short)0, acc, false, false);
    float2 c2 = w2p[kk], d2 = hp[kk];
    v2f c; c.x = c2.x; c.y = c2.y;
    v2f dd; dd.x = d2.x; dd.y = d2.y;
    acc = __builtin_amdgcn_wmma_f32_16x16x4_f32(false, c, false, dd,
                                                (short)0, acc, false, false);
  }
  if (m == 0) {
    const int base = tile * 16 + hi * 8;
    #pragma unroll
    for (int v = 0; v < 8; ++v)
      out[base + v] = tanhf(acc[v] + b1[base + v] + b2[base + v]);
  }
}

// ---------------------------------------------------------------------------
// launch
// ---------------------------------------------------------------------------
extern "C" void kernel_launch(void* const* d_in, const int* in_sizes, int n_in,
                              void* d_out, int out_size, void* d_ws, size_t ws_size,
                              hipStream_t stream) {
  const float* H  = (const float*)d_in[0];
  const float* ht = (const float*)d_in[1];
  const float* W0 = (const float*)d_in[2];
  const float* b0 = (const float*)d_in[3];
  const float* W1 = (const float*)d_in[4];
  const float* b1 = (const float*)d_in[5];
  const float* W2 = (const float*)d_in[6];
  const float* b2 = (const float*)d_in[7];
  float* out = (float*)d_out;

  const int n = in_sizes[0] / HD;          // N_NODES = 50000

  // workspace layout (floats)
  float* ws      = (float*)d_ws;
  float* q       = ws;                     // 1024
  float* s       = q + HD;                 // n
  float* w       = s + n;                  // n
  float* es_part = w + n;                  // NCHUNK*1024
  float* es      = es_part + NCHUNK * HD;  // 1024
  float* denom   = es + HD;                // 1
  unsigned* mb   = (unsigned*)(denom + 1); // 1

  const int rowsPerChunk = (n + NCHUNK - 1) / NCHUNK;

  kInit<<<1, 1, 0, stream>>>(mb);
  kMatvecWmma<<<HD / 16, 32, 0, stream>>>(W0, ht, b0, q);
  kScores<<<(n + 7) / 8, 256, 0, stream>>>(H, q, s, mb, n);
  kDenom<<<1, 1024, 0, stream>>>(s, w, mb, denom, n);
  kWeightedSum<<<NCHUNK * 4, 256, 0, stream>>>(H, w, es_part, n, rowsPerChunk);
  kReduceEs<<<HD / 256, 256, 0, stream>>>(es_part, denom, es);
  kFinalWmma<<<HD / 16, 32, 0, stream>>>(W1, b1, W2, b2, es, ht, out);
}